// MixerModelEmbedding_3332894621876
// MI455X (gfx1250) — compile-verified
//
#include <hip/hip_runtime.h>

// Embedding gather: out[b][d][l] = (x[l][b] < 20 ? W[x[l][b]][d] : 0) * sqrt(8)
// Pure store-bandwidth-bound problem (128 MiB out, 80 KB live table).
// CDNA5 path used: async global->LDS staging (ASYNCcnt) + b128 stores.

#define VOCAB      50257
#define D_MODEL    1024
#define SEQ_LEN    2048
#define BATCH      16
#define MAX_TOK    20
#define MUP_SCALE  2.8284271247461903f   // sqrt(8)

#define ROW_STRIDE 1025                  // padded LDS row stride (dwords) -> bank spread
#define L_TILE     128                   // l-values per block
#define NTHREADS   256

__global__ __launch_bounds__(NTHREADS)
void embed_gather_kernel(const int* __restrict__ x,
                         const float* __restrict__ w,
                         float* __restrict__ out)
{
    // rows 0..19 = live embedding rows, row 20 = zeros (clamp target)
    __shared__ float tab[(MAX_TOK + 1) * ROW_STRIDE];
    __shared__ int   tok_s[L_TILE];

    const int t      = threadIdx.x;
    const int blk    = blockIdx.x;
    const int b      = blk >> 4;               // 16 l-tiles per batch entry
    const int l_tile = (blk & 15) * L_TILE;

    // ---- zero the clamp row (row 20): 256 threads x 4 floats = 1024 ----
    {
        float* z = &tab[MAX_TOK * ROW_STRIDE + t * 4];
        z[0] = 0.0f; z[1] = 0.0f; z[2] = 0.0f; z[3] = 0.0f;
    }

    // ---- stage live table rows 0..19 into LDS via async global->LDS b128 ----
    // 20 rows * 64 chunks(16B) = 1280 chunks = 5 per thread.
    {
        const unsigned long long wbase = (unsigned long long)(uintptr_t)w;
#pragma unroll
        for (int it = 0; it < 5; ++it) {
            int chunk = it * NTHREADS + t;            // 0..1279
            int row   = chunk >> 6;                   // 0..19
            int c16   = chunk & 63;                   // 16B chunk within row
            unsigned goff = (unsigned)((row * D_MODEL    + c16 * 4) * 4);
            unsigned loff = (unsigned)(uintptr_t)(&tab[row * ROW_STRIDE + c16 * 4]);
            // GVS mode: mem_addr = SADDR64 + VADDR32 ; VDST = LDS byte address
            asm volatile("global_load_async_to_lds_b128 %0, %1, %2"
                         :: "v"(loff), "v"(goff), "s"(wbase)
                         : "memory");
        }
    }

    // ---- stage the 128 tokens for this (b, l-tile) ----
    if (t < L_TILE)
        tok_s[t] = x[(l_tile + t) * BATCH + b];

    // fence async staging for this wave, then sync the workgroup
    asm volatile("s_wait_asynccnt 0" ::: "memory");
    __syncthreads();

    // ---- main loop: lane -> 4 consecutive l, wave covers 128 contiguous l ----
    const int lane_l = t & 31;                 // 0..31  -> l4 = tile + lane_l*4
    const int d0     = t >> 5;                 // 0..7   -> d = d0 + 8k
    const int l4     = l_tile + lane_l * 4;

    int base0, base1, base2, base3;
    {
        int t0 = tok_s[lane_l * 4 + 0];
        int t1 = tok_s[lane_l * 4 + 1];
        int t2 = tok_s[lane_l * 4 + 2];
        int t3 = tok_s[lane_l * 4 + 3];
        base0 = ((t0 < MAX_TOK) ? t0 : MAX_TOK) * ROW_STRIDE;
        base1 = ((t1 < MAX_TOK) ? t1 : MAX_TOK) * ROW_STRIDE;
        base2 = ((t2 < MAX_TOK) ? t2 : MAX_TOK) * ROW_STRIDE;
        base3 = ((t3 < MAX_TOK) ? t3 : MAX_TOK) * ROW_STRIDE;
    }

    float* outp = out + ((size_t)(b * D_MODEL + d0)) * SEQ_LEN + l4;

#pragma unroll 4
    for (int k = 0; k < D_MODEL / 8; ++k) {
        const int d = d0 + k * 8;
        float4 v;
        v.x = tab[base0 + d] * MUP_SCALE;
        v.y = tab[base1 + d] * MUP_SCALE;
        v.z = tab[base2 + d] * MUP_SCALE;
        v.w = tab[base3 + d] * MUP_SCALE;
        *reinterpret_cast<float4*>(outp) = v;   // b128 store, 512B contiguous per wave
        outp += 8 * SEQ_LEN;
    }
}

extern "C" void kernel_launch(void* const* d_in, const int* in_sizes, int n_in,
                              void* d_out, int out_size, void* d_ws, size_t ws_size,
                              hipStream_t stream)
{
    const int*   x = (const int*)d_in[0];     // (SEQ_LEN, BATCH) int32 token ids
    const float* w = (const float*)d_in[1];   // (VOCAB, D_MODEL) fp32
    float*     out = (float*)d_out;           // (BATCH, D_MODEL, SEQ_LEN) fp32

    dim3 grid(BATCH * (SEQ_LEN / L_TILE));    // 16 * 16 = 256 blocks
    dim3 block(NTHREADS);
    embed_gather_kernel<<<grid, block, 0, stream>>>(x, w, out);
}